// MemoryEfficientSAM2Attention3D_1657857376469
// MI455X (gfx1250) — compile-verified
//
#include <hip/hip_runtime.h>
#include <math.h>

typedef __attribute__((ext_vector_type(2))) float v2f;
typedef __attribute__((ext_vector_type(8))) float v8f;

// D = A(16x4 f32) * B(4x16 f32) + C(16x16 f32), full wave32 op.
#define WMMA_F32(a, b, c) \
  __builtin_amdgcn_wmma_f32_16x16x4_f32(false, (a), false, (b), (short)0, (c), false, false)

#define NEG_HUGE (-3.0e38f)

// ---------------------------------------------------------------------------
// K1: xp[b,hp,wp,dp,c] = mean over 4x4x4 block of x.  (pool commutes with the
// 1x1 convs, so pooling first shrinks all GEMMs by 64x.)
// threads: B*16*16*16*32 = 262144, each handles 4 channels (float4).
// ---------------------------------------------------------------------------
__global__ void sam3d_pool(const float* __restrict__ x, float* __restrict__ xp) {
  int t = blockIdx.x * blockDim.x + threadIdx.x;
  int c4 = t & 31;
  int dp = (t >> 5) & 15;
  int wp = (t >> 9) & 15;
  int hp = (t >> 13) & 15;
  int b  = t >> 17;
  const float4* X = (const float4*)x;
  float sx = 0.f, sy = 0.f, sz = 0.f, sw = 0.f;
  for (int i = 0; i < 4; ++i)
    for (int j = 0; j < 4; ++j)
      for (int k = 0; k < 4; ++k) {
        int idx = (((b * 64 + hp * 4 + i) * 64 + wp * 4 + j) * 64 + dp * 4 + k) * 32 + c4;
        float4 v = X[idx];
        sx += v.x; sy += v.y; sz += v.z; sw += v.w;
      }
  const float inv = 1.0f / 64.0f;
  float4 o; o.x = sx * inv; o.y = sy * inv; o.z = sz * inv; o.w = sw * inv;
  ((float4*)xp)[(((b * 16 + hp) * 16 + wp) * 16 + dp) * 32 + c4] = o;
}

// ---------------------------------------------------------------------------
// K2: one wave (32 threads) per 16-row tile.
//  tiles 0..511  : rows s=(tile%256)*16..+15 -> kf (16x16) and vf (16x128)
//  tiles 512..543: compact q rows s=16*j (only dp==0 queries are ever used)
// K=128 reduced with 32 x V_WMMA_F32_16X16X4_F32 per output N-tile.
// A and B fragments are fully preloaded into registers so the 32 WMMAs issue
// back-to-back (no per-WMMA s_wait_loadcnt round trips).
// Softmax scale 1/sqrt(E)=0.25 folded into q.
// ---------------------------------------------------------------------------
__global__ void sam3d_qkv(const float* __restrict__ xp,
                          const float* __restrict__ Wq, const float* __restrict__ bq,
                          const float* __restrict__ Wk, const float* __restrict__ bk,
                          const float* __restrict__ Wv, const float* __restrict__ bv,
                          float* __restrict__ kf, float* __restrict__ vf,
                          float* __restrict__ qc) {
  int tile = blockIdx.x;
  int lane = threadIdx.x;
  int half = lane >> 4;      // 0: K={0,1}+4k, 1: K={2,3}+4k (A/B lane split)
  int lm   = lane & 15;      // A row m / B,C,D column n

  bool isQ = tile >= 512;
  int b, rowbase, rowstride;
  if (!isQ) { b = tile >> 8; rowbase = b * 4096 + (tile & 255) * 16; rowstride = 1; }
  else      { int jt = tile - 512; b = jt >> 4; rowbase = b * 4096 + (jt & 15) * 256; rowstride = 16; }

  // Cache the 16x128 A tile: per lane, row m=lm, 2 f32 per K-step (64 regs).
  float a[64];
  const float* arow = xp + (rowbase + rowstride * lm) * 128;
#pragma unroll
  for (int kb = 0; kb < 32; ++kb) {
    a[2 * kb]     = arow[4 * kb + 2 * half];
    a[2 * kb + 1] = arow[4 * kb + 2 * half + 1];
  }

  if (isQ) {
    float bw[64];
#pragma unroll
    for (int kb = 0; kb < 32; ++kb) {
      bw[2 * kb]     = Wq[(4 * kb + 2 * half) * 16 + lm];
      bw[2 * kb + 1] = Wq[(4 * kb + 2 * half + 1) * 16 + lm];
    }
    v8f acc = {};
#pragma unroll
    for (int kb = 0; kb < 32; ++kb) {
      v2f av; av.x = a[2 * kb];  av.y = a[2 * kb + 1];
      v2f bb; bb.x = bw[2 * kb]; bb.y = bw[2 * kb + 1];
      acc = WMMA_F32(av, bb, acc);
    }
    float bias = bq[lm];
    int jt = tile - 512; int j16 = jt & 15;
#pragma unroll
    for (int r = 0; r < 8; ++r) {
      int m = r + 8 * half;
      qc[(b * 256 + j16 * 16 + m) * 16 + lm] = (acc[r] + bias) * 0.25f;
    }
    return;
  }

  // k projection: N-tile of 16 (E=16)
  {
    float bw[64];
#pragma unroll
    for (int kb = 0; kb < 32; ++kb) {
      bw[2 * kb]     = Wk[(4 * kb + 2 * half) * 16 + lm];
      bw[2 * kb + 1] = Wk[(4 * kb + 2 * half + 1) * 16 + lm];
    }
    v8f acc = {};
#pragma unroll
    for (int kb = 0; kb < 32; ++kb) {
      v2f av; av.x = a[2 * kb];  av.y = a[2 * kb + 1];
      v2f bb; bb.x = bw[2 * kb]; bb.y = bw[2 * kb + 1];
      acc = WMMA_F32(av, bb, acc);
    }
    float bias = bk[lm];
#pragma unroll
    for (int r = 0; r < 8; ++r) {
      int m = r + 8 * half;
      kf[(rowbase + m) * 16 + lm] = acc[r] + bias;
    }
  }
  // v projection: 8 N-tiles of 16 (C=128)
  for (int nt = 0; nt < 8; ++nt) {
    float bw[64];
#pragma unroll
    for (int kb = 0; kb < 32; ++kb) {
      bw[2 * kb]     = Wv[(4 * kb + 2 * half) * 128 + nt * 16 + lm];
      bw[2 * kb + 1] = Wv[(4 * kb + 2 * half + 1) * 128 + nt * 16 + lm];
    }
    v8f acc = {};
#pragma unroll
    for (int kb = 0; kb < 32; ++kb) {
      v2f av; av.x = a[2 * kb];  av.y = a[2 * kb + 1];
      v2f bb; bb.x = bw[2 * kb]; bb.y = bw[2 * kb + 1];
      acc = WMMA_F32(av, bb, acc);
    }
    float bias = bv[nt * 16 + lm];
#pragma unroll
    for (int r = 0; r < 8; ++r) {
      int m = r + 8 * half;
      vf[(rowbase + m) * 128 + nt * 16 + lm] = acc[r] + bias;
    }
  }
}

// ---------------------------------------------------------------------------
// K3: flash attention for the 256 compact queries per batch.
// grid = 32 (b, qtile). block = 128 (4 waves); each wave handles 1024 keys
// (64 key-tiles) with online softmax; partials merged in LDS (log-sum-exp).
// K and V fragments for the tile are preloaded at the top of each iteration
// so their latency overlaps the score WMMAs and softmax VALU work.
// ---------------------------------------------------------------------------
__global__ void sam3d_attn(const float* __restrict__ qc,
                           const float* __restrict__ kf,
                           const float* __restrict__ vf,
                           float* __restrict__ att) {
  __shared__ float ldsP[4][256];        // 16x16 P staging per wave (D->A relayout)
  __shared__ float ldsAcc[4][16][128];  // per-wave partial attended tiles
  __shared__ float ldsRM[4][16];        // per-wave row max
  __shared__ float ldsLS[4][16];        // per-wave row sum

  int tid  = threadIdx.x;
  int wave = tid >> 5;
  int lane = tid & 31;
  int half = lane >> 4;
  int lm   = lane & 15;
  int b  = blockIdx.x >> 4;
  int qt = blockIdx.x & 15;

  // q A-fragments for all 4 K-steps (E=16), loaded once.
  float qa[8];
  {
    const float* qrow = qc + (b * 256 + qt * 16 + lm) * 16;
#pragma unroll
    for (int st = 0; st < 4; ++st) {
      qa[2 * st]     = qrow[4 * st + 2 * half];
      qa[2 * st + 1] = qrow[4 * st + 2 * half + 1];
    }
  }

  v8f acc[8];
#pragma unroll
  for (int nt = 0; nt < 8; ++nt) acc[nt] = (v8f){};
  float rm[8], ls[8];
#pragma unroll
  for (int r = 0; r < 8; ++r) { rm[r] = NEG_HUGE; ls[r] = 0.f; }

  const float* kfb = kf + b * 4096 * 16;
  const float* vfb = vf + b * 4096 * 128;

  for (int it = 0; it < 64; ++it) {
    int t0 = (wave * 64 + it) * 16;

    // ---- issue ALL tile loads first (k: 8 floats, v: 64 floats per lane) ----
    float kb8[8];
#pragma unroll
    for (int st = 0; st < 4; ++st) {
      const float* kr = kfb + (t0 + lm) * 16 + 4 * st + 2 * half;
      kb8[2 * st]     = kr[0];
      kb8[2 * st + 1] = kr[1];
    }
    float varr[64];   // [st][nt][2] : B-fragments of the 16x128 V tile
#pragma unroll
    for (int st = 0; st < 4; ++st) {
      const float* vr = vfb + (t0 + 4 * st + 2 * half) * 128 + lm;
#pragma unroll
      for (int nt = 0; nt < 8; ++nt) {
        varr[st * 16 + 2 * nt]     = vr[nt * 16];
        varr[st * 16 + 2 * nt + 1] = vr[nt * 16 + 128];
      }
    }

    // scores tile S = q (16xE) * k^T (Ex16), E=16 -> 4 WMMA steps
    v8f s = {};
#pragma unroll
    for (int st = 0; st < 4; ++st) {
      v2f av; av.x = qa[2 * st];  av.y = qa[2 * st + 1];
      v2f bb; bb.x = kb8[2 * st]; bb.y = kb8[2 * st + 1];
      s = WMMA_F32(av, bb, s);
    }

    // online softmax; row reductions stay inside each 16-lane half.
#pragma unroll
    for (int r = 0; r < 8; ++r) {
      float mx = s[r];
      mx = fmaxf(mx, __shfl_xor(mx, 1, 32));
      mx = fmaxf(mx, __shfl_xor(mx, 2, 32));
      mx = fmaxf(mx, __shfl_xor(mx, 4, 32));
      mx = fmaxf(mx, __shfl_xor(mx, 8, 32));
      float nm    = fmaxf(rm[r], mx);
      float alpha = __expf(rm[r] - nm);
      float p     = __expf(s[r] - nm);
      float rs = p;
      rs += __shfl_xor(rs, 1, 32);
      rs += __shfl_xor(rs, 2, 32);
      rs += __shfl_xor(rs, 4, 32);
      rs += __shfl_xor(rs, 8, 32);
      ls[r] = ls[r] * alpha + rs;
      rm[r] = nm;
#pragma unroll
      for (int nt = 0; nt < 8; ++nt) acc[nt][r] *= alpha;
      int m = r + 8 * half;
      ldsP[wave][m * 16 + lm] = p;   // D-layout -> row-major 16x16
    }
    // cross-lane LDS hop within the wave: drain DS before consuming.
    asm volatile("s_wait_dscnt 0x0" ::: "memory");

    // attended += P (16x16) * V (16x128): 4 K-steps x 8 N-tiles, all-register B
#pragma unroll
    for (int st = 0; st < 4; ++st) {
      v2f pa; pa.x = ldsP[wave][lm * 16 + 4 * st + 2 * half];
              pa.y = ldsP[wave][lm * 16 + 4 * st + 2 * half + 1];
#pragma unroll
      for (int nt = 0; nt < 8; ++nt) {
        v2f bb; bb.x = varr[st * 16 + 2 * nt]; bb.y = varr[st * 16 + 2 * nt + 1];
        acc[nt] = WMMA_F32(pa, bb, acc[nt]);
      }
    }
  }

  // stage per-wave partials
#pragma unroll
  for (int r = 0; r < 8; ++r) {
    int m = r + 8 * half;
#pragma unroll
    for (int nt = 0; nt < 8; ++nt) ldsAcc[wave][m][nt * 16 + lm] = acc[nt][r];
    if (lm == 0) { ldsRM[wave][m] = rm[r]; ldsLS[wave][m] = ls[r]; }
  }
  __syncthreads();

  // merge 4 partials (log-sum-exp) and normalize; 2048 outputs / 128 threads
  for (int o = tid; o < 2048; o += 128) {
    int row = o >> 7, c = o & 127;
    float M = ldsRM[0][row];
    for (int w = 1; w < 4; ++w) M = fmaxf(M, ldsRM[w][row]);
    float denom = 0.f, num = 0.f;
    for (int w = 0; w < 4; ++w) {
      float f = __expf(ldsRM[w][row] - M);
      denom += f * ldsLS[w][row];
      num   += f * ldsAcc[w][row][c];
    }
    att[(b * 256 + qt * 16 + row) * 128 + c] = num / denom;
  }
}

// ---------------------------------------------------------------------------
// K4: out = x + gamma * bilinear_upsample(att)[h,w] broadcast over d.
// jax.image.resize 'linear': src = (dst+0.5)/4 - 0.5, indices edge-clamped.
// threads: B*64*64*32 = 524288; each covers 4 channels and loops d (64).
// ---------------------------------------------------------------------------
__global__ void sam3d_upsample_add(const float* __restrict__ x,
                                   const float* __restrict__ att,
                                   const float* __restrict__ gamma,
                                   float* __restrict__ out) {
  int t = blockIdx.x * blockDim.x + threadIdx.x;
  int c4 = t & 31;
  int w  = (t >> 5) & 63;
  int h  = (t >> 11) & 63;
  int b  = t >> 17;

  float fy = 0.25f * (float)h - 0.375f;
  float fx = 0.25f * (float)w - 0.375f;
  float y0f = floorf(fy), x0f = floorf(fx);
  float ty = fy - y0f, tx = fx - x0f;
  int iy0 = min(max((int)y0f, 0), 15), iy1 = min(max((int)y0f + 1, 0), 15);
  int ix0 = min(max((int)x0f, 0), 15), ix1 = min(max((int)x0f + 1, 0), 15);

  const float4* A = (const float4*)att;
  float4 a00 = A[(b * 256 + iy0 * 16 + ix0) * 32 + c4];
  float4 a01 = A[(b * 256 + iy0 * 16 + ix1) * 32 + c4];
  float4 a10 = A[(b * 256 + iy1 * 16 + ix0) * 32 + c4];
  float4 a11 = A[(b * 256 + iy1 * 16 + ix1) * 32 + c4];
  float w00 = (1.f - ty) * (1.f - tx), w01 = (1.f - ty) * tx;
  float w10 = ty * (1.f - tx),         w11 = ty * tx;
  float g = gamma[0];
  float upx = g * (w00 * a00.x + w01 * a01.x + w10 * a10.x + w11 * a11.x);
  float upy = g * (w00 * a00.y + w01 * a01.y + w10 * a10.y + w11 * a11.y);
  float upz = g * (w00 * a00.z + w01 * a01.z + w10 * a10.z + w11 * a11.z);
  float upw = g * (w00 * a00.w + w01 * a01.w + w10 * a10.w + w11 * a11.w);

  const float4* X = (const float4*)x;
  float4* O = (float4*)out;
  int base = (((b * 64 + h) * 64 + w) * 64) * 32 + c4;
  for (int d = 0; d < 64; ++d) {
    int idx = base + d * 32;
    float4 xv = X[idx];
    float4 ov; ov.x = xv.x + upx; ov.y = xv.y + upy; ov.z = xv.z + upz; ov.w = xv.w + upw;
    O[idx] = ov;
  }
}

extern "C" void kernel_launch(void* const* d_in, const int* in_sizes, int n_in,
                              void* d_out, int out_size, void* d_ws, size_t ws_size,
                              hipStream_t stream) {
  const float* x     = (const float*)d_in[0];
  const float* Wq    = (const float*)d_in[1];
  const float* bq    = (const float*)d_in[2];
  const float* Wk    = (const float*)d_in[3];
  const float* bk    = (const float*)d_in[4];
  const float* Wv    = (const float*)d_in[5];
  const float* bv    = (const float*)d_in[6];
  const float* gamma = (const float*)d_in[7];
  float* out = (float*)d_out;

  // workspace layout (floats): ~9.2 MB total
  float* ws  = (float*)d_ws;
  float* xp  = ws;               // B*4096*128 = 1048576
  float* kf  = xp + 1048576;     // B*4096*16  =  131072
  float* vf  = kf + 131072;      // B*4096*128 = 1048576
  float* qc  = vf + 1048576;     // B*256*16   =    8192
  float* att = qc + 8192;        // B*256*128  =   65536

  sam3d_pool<<<1024, 256, 0, stream>>>(x, xp);
  sam3d_qkv<<<544, 32, 0, stream>>>(xp, Wq, bq, Wk, bk, Wv, bv, kf, vf, qc);
  sam3d_attn<<<32, 128, 0, stream>>>(qc, kf, vf, att);
  sam3d_upsample_add<<<2048, 256, 0, stream>>>(x, att, gamma, out);
}